// DHYPR_15745350107691
// MI455X (gfx1250) — compile-verified
//
#include <hip/hip_runtime.h>
#include <hip/hip_bf16.h>

typedef __attribute__((ext_vector_type(2))) float v2f;
typedef __attribute__((ext_vector_type(8))) float v8f;

#define MIN_NORM 1e-15f
#define MAXN     (1.0f - 4e-3f)   // (1-EPS)/sqrt(c), c=1

#define N_NODES  50000
#define FEAT     128
#define HID      64
#define DIM      32
#define NBRANCH  8
#define NEDGE    800000

// ---------------- helpers ----------------

__device__ __forceinline__ float warp_sum(float v) {
#pragma unroll
    for (int o = 16; o > 0; o >>= 1) v += __shfl_xor(v, o, 32);
    return v;
}

__device__ __forceinline__ float art(float x) {
    x = fminf(fmaxf(x, -1.0f + 1e-7f), 1.0f - 1e-7f);
    return atanhf(x);
}

// ---------------- K0: x_hyp = proj(expmap0(x)), store row norm ----------------

__global__ void k_expmap_in(const float* __restrict__ x, float* __restrict__ xh,
                            float* __restrict__ xn, int M) {
    int row  = blockIdx.x * (blockDim.x >> 5) + (threadIdx.x >> 5);
    int lane = threadIdx.x & 31;
    if (row >= M) return;
    const float* xr = x + (size_t)row * FEAT;
    float v[4]; float s = 0.0f;
#pragma unroll
    for (int i = 0; i < 4; i++) { v[i] = xr[lane + 32 * i]; s += v[i] * v[i]; }
    s = warp_sum(s);
    float n  = fmaxf(sqrtf(s), MIN_NORM);
    float t  = tanhf(n);
    float sc = t / n;
    float nn = fmaxf(t, MIN_NORM);
    if (nn > MAXN) { sc *= MAXN / nn; nn = MAXN; }
    float* yr = xh + (size_t)row * FEAT;
#pragma unroll
    for (int i = 0; i < 4; i++) yr[lane + 32 * i] = v[i] * sc;
    if (lane == 0) xn[row] = nn;
}

// ---------------- bias: hb = proj(expmap0(b)) (one wave) ----------------

template <int D>
__global__ void k_bias(const float* __restrict__ b, float* __restrict__ hb) {
    constexpr int P = D / 32;
    int lane = threadIdx.x & 31;
    float v[P]; float s = 0.0f;
#pragma unroll
    for (int i = 0; i < P; i++) { v[i] = b[lane + 32 * i]; s += v[i] * v[i]; }
    s = warp_sum(s);
    float n  = fmaxf(sqrtf(s), MIN_NORM);
    float t  = tanhf(n);
    float sc = t / n;
    float nn = fmaxf(t, MIN_NORM);
    if (nn > MAXN) sc *= MAXN / nn;
#pragma unroll
    for (int i = 0; i < P; i++) hb[lane + 32 * i] = v[i] * sc;
}

// ---------------- GEMM: Y[M,N] = X[M,K] @ W[N,K]^T via V_WMMA_F32_16X16X4 ----------------
// One wave owns a 16-row tile of X and produces all N columns; A fragment is
// reused across the N/16 column tiles (WMMA A-reuse hint left off: operands differ).

template <int K, int N>
__global__ void k_gemm_wmma(const float* __restrict__ X, const float* __restrict__ W,
                            float* __restrict__ Y, int M) {
    int wave = (int)((blockIdx.x * blockDim.x + threadIdx.x) >> 5);
    int lane = threadIdx.x & 31;
    int m0 = wave * 16;
    if (m0 >= M) return;
    int half = lane >> 4;
    int l    = lane & 15;
    constexpr int NT = N / 16;

    v8f acc[NT];
#pragma unroll
    for (int t = 0; t < NT; t++)
#pragma unroll
        for (int j = 0; j < 8; j++) acc[t][j] = 0.0f;

    const float* xr = X + (size_t)(m0 + l) * K;
#pragma unroll
    for (int k0 = 0; k0 < K; k0 += 4) {
        // A 16x4 layout: v0 = K=k0 (lanes 0-15) / K=k0+2 (lanes 16-31); v1 = +1
        int ka = k0 + 2 * half;
        v2f a; a.x = xr[ka]; a.y = xr[ka + 1];
#pragma unroll
        for (int t = 0; t < NT; t++) {
            // B 4x16: B[k][n] = W[n][k], n striped across lanes (mirrors A halves)
            const float* wr = W + (size_t)(t * 16 + l) * K;
            v2f b; b.x = wr[ka]; b.y = wr[ka + 1];
            acc[t] = __builtin_amdgcn_wmma_f32_16x16x4_f32(
                false, a, false, b, (short)0, acc[t], false, false);
        }
    }
    // C/D layout: VGPR j -> row m0 + j + 8*half, col n0 + l
#pragma unroll
    for (int t = 0; t < NT; t++)
#pragma unroll
        for (int j = 0; j < 8; j++)
            Y[(size_t)(m0 + j + 8 * half) * N + t * 16 + l] = acc[t][j];
}

// ---------------- HypLinear tail: mobius rescale + bias mobius_add + logmap0 ----------------

template <int D>
__global__ void k_hyplin_fix(float* __restrict__ mx, const float* __restrict__ xnorm,
                             const float* __restrict__ hb, int M) {
    constexpr int P = D / 32;
    int row  = blockIdx.x * (blockDim.x >> 5) + (threadIdx.x >> 5);
    int lane = threadIdx.x & 31;
    if (row >= M) return;
    float* mr = mx + (size_t)row * D;
    float m[P], bb[P]; float sm = 0.0f;
#pragma unroll
    for (int i = 0; i < P; i++) {
        m[i] = mr[lane + 32 * i];
        bb[i] = hb[lane + 32 * i];
        sm += m[i] * m[i];
    }
    sm = warp_sum(sm);
    float mxn = fmaxf(sqrtf(sm), MIN_NORM);
    float xr  = fmaxf(xnorm[row], MIN_NORM);
    float th  = tanhf(mxn / xr * art(xr));     // mobius_matvec scale, c=1
    float sc  = th / mxn;
    float hn  = fmaxf(fabsf(th), MIN_NORM);
    if (hn > MAXN) { sc *= MAXN / hn; hn = MAXN; }   // proj
    float h[P]; float dxy = 0.0f, dyy = 0.0f;
#pragma unroll
    for (int i = 0; i < P; i++) {
        h[i] = m[i] * sc;
        dxy += h[i] * bb[i];
        dyy += bb[i] * bb[i];
    }
    float xy = warp_sum(dxy);
    float y2 = warp_sum(dyy);
    float x2 = hn * hn;
    float den = fmaxf(1.0f + 2.0f * xy + x2 * y2, MIN_NORM);
    float ca = (1.0f + 2.0f * xy + y2) / den;
    float cb = (1.0f - x2) / den;
    float r[P]; float sr = 0.0f;
#pragma unroll
    for (int i = 0; i < P; i++) { r[i] = ca * h[i] + cb * bb[i]; sr += r[i] * r[i]; }
    sr = warp_sum(sr);
    float rn = fmaxf(sqrtf(sr), MIN_NORM);
    float ps = (rn > MAXN) ? (MAXN / rn) : 1.0f;    // proj
    float rc = fminf(rn, MAXN);
    float ls = art(rc) / rc * ps;                    // logmap0 scale
#pragma unroll
    for (int i = 0; i < P; i++) mr[lane + 32 * i] = ls * r[i];
}

// ---------------- SpMM scatter: agg[row] += val * xt[col] ----------------

template <int D>
__global__ void k_spmm(const int* __restrict__ rows, const int* __restrict__ cols,
                       const float* __restrict__ vals, const float* __restrict__ xt,
                       float* __restrict__ agg, int nE) {
    constexpr int CH = D / 4;
    long t = (long)blockIdx.x * blockDim.x + threadIdx.x;
    int e  = (int)(t / CH);
    int c4 = (int)(t % CH) * 4;
    if (e >= nE) return;
    int r = rows[e], c = cols[e];
    float v = vals[e];
    const float4 s = *(const float4*)(xt + (size_t)c * D + c4);
    float* dst = agg + (size_t)r * D + c4;
    atomicAdd(dst + 0, v * s.x);
    atomicAdd(dst + 1, v * s.y);
    atomicAdd(dst + 2, v * s.z);
    atomicAdd(dst + 3, v * s.w);
}

// ---------------- layer-1 post-agg: h1 = proj(expmap0(relu(logmap0(proj(expmap0(agg)))))) ----------------

__global__ void k_postagg1(const float* __restrict__ agg, float* __restrict__ h1,
                           float* __restrict__ h1n, int M) {
    int row  = blockIdx.x * (blockDim.x >> 5) + (threadIdx.x >> 5);
    int lane = threadIdx.x & 31;
    if (row >= M) return;
    const float* ar = agg + (size_t)row * HID;
    float a0 = ar[lane], a1 = ar[lane + 32];
    float n2 = warp_sum(a0 * a0 + a1 * a1);
    float n  = fmaxf(sqrtf(n2), MIN_NORM);
    float t  = tanhf(n);
    float sc = t / n;
    float hn = fmaxf(t, MIN_NORM);
    if (hn > MAXN) { sc *= MAXN / hn; hn = MAXN; }
    float ls = art(hn) / hn;
    float v0 = fmaxf(ls * a0 * sc, 0.0f);
    float v1 = fmaxf(ls * a1 * sc, 0.0f);
    float vn2 = warp_sum(v0 * v0 + v1 * v1);
    float vn = fmaxf(sqrtf(vn2), MIN_NORM);
    float tv = tanhf(vn);
    float sv = tv / vn;
    float bn = fmaxf(tv, MIN_NORM);
    if (bn > MAXN) { sv *= MAXN / bn; bn = MAXN; }
    float* hr = h1 + (size_t)row * HID;
    hr[lane] = v0 * sv;
    hr[lane + 32] = v1 * sv;
    if (lane == 0) h1n[row] = bn;
}

// ---------------- layer-2 post-agg + mobius fold + tangent sum (dim 32, lane==dim) ----------------

__global__ void k_post2_combine(const float* __restrict__ agg, float* __restrict__ tgt,
                                float* __restrict__ tsum, int ib, int M) {
    int row  = blockIdx.x * (blockDim.x >> 5) + (threadIdx.x >> 5);
    int lane = threadIdx.x & 31;
    if (row >= M) return;
    size_t idx = (size_t)row * DIM + lane;
    float a = agg[idx];
    // h = proj(expmap0(agg))
    float n2 = warp_sum(a * a);
    float n  = fmaxf(sqrtf(n2), MIN_NORM);
    float t  = tanhf(n);
    float sc = t / n;
    float hn = fmaxf(t, MIN_NORM);
    if (hn > MAXN) { sc *= MAXN / hn; hn = MAXN; }
    // v = relu(logmap0(h)); bo = proj(expmap0(v))
    float v  = fmaxf(art(hn) / hn * a * sc, 0.0f);
    float vn2 = warp_sum(v * v);
    float vn = fmaxf(sqrtf(vn2), MIN_NORM);
    float tv = tanhf(vn);
    float sv = tv / vn;
    float bn = fmaxf(tv, MIN_NORM);
    if (bn > MAXN) { sv *= MAXN / bn; bn = MAXN; }
    float bo = v * sv;
    // tansum += logmap0(bo)
    tsum[idx] += art(bn) / bn * bo;
    // w = mobius_mulscaler(1/8, bo)
    float wmag = tanhf(0.125f * art(bn));
    float w = bo * (wmag / bn);
    if (ib == 0) { tgt[idx] = w; return; }
    // target = mobius_add(target, w)
    float xo = tgt[idx];
    float x2 = warp_sum(xo * xo);
    float xy = warp_sum(xo * w);
    float y2 = wmag * wmag;
    float den = fmaxf(1.0f + 2.0f * xy + x2 * y2, MIN_NORM);
    float ca = (1.0f + 2.0f * xy + y2) / den;
    float cb = (1.0f - x2) / den;
    tgt[idx] = ca * xo + cb * w;
}

// ---------------- final: out = proj(expmap0((tsum + logmap0(target)) / 9)) ----------------

__global__ void k_final(const float* __restrict__ tsum, const float* __restrict__ tgt,
                        float* __restrict__ out, int M) {
    int row  = blockIdx.x * (blockDim.x >> 5) + (threadIdx.x >> 5);
    int lane = threadIdx.x & 31;
    if (row >= M) return;
    size_t idx = (size_t)row * DIM + lane;
    float tg = tgt[idx];
    float tn2 = warp_sum(tg * tg);
    float tn = fmaxf(sqrtf(tn2), MIN_NORM);
    float r0 = (tsum[idx] + art(tn) / tn * tg) / 9.0f;
    float rn2 = warp_sum(r0 * r0);
    float rn = fmaxf(sqrtf(rn2), MIN_NORM);
    float t  = tanhf(rn);
    float sc = t / rn;
    float on = fmaxf(t, MIN_NORM);
    if (on > MAXN) sc *= MAXN / on;
    out[idx] = r0 * sc;
}

// ---------------- launch ----------------

extern "C" void kernel_launch(void* const* d_in, const int* in_sizes, int n_in,
                              void* d_out, int out_size, void* d_ws, size_t ws_size,
                              hipStream_t stream) {
    const float* x      = (const float*)d_in[0];
    const int*   k_rows = (const int*)d_in[1];
    const int*   k_cols = (const int*)d_in[2];
    const float* k_vals = (const float*)d_in[3];
    const float* W1     = (const float*)d_in[4];
    const float* b1     = (const float*)d_in[5];
    const float* W2     = (const float*)d_in[6];
    const float* b2     = (const float*)d_in[7];

    // workspace layout (floats), ~90 MB total
    float* ws   = (float*)d_ws;
    float* xh   = ws;                               // 50000*128
    float* xn   = xh   + (size_t)N_NODES * FEAT;    // 50000
    float* mx1  = xn   + N_NODES;                   // 50000*64 (mx -> xt, in place)
    float* agg1 = mx1  + (size_t)N_NODES * HID;     // 50000*64
    float* h1   = agg1 + (size_t)N_NODES * HID;     // 50000*64
    float* h1n  = h1   + (size_t)N_NODES * HID;     // 50000
    float* mx2  = h1n  + N_NODES;                   // 50000*32
    float* agg2 = mx2  + (size_t)N_NODES * DIM;     // 50000*32
    float* tgt  = agg2 + (size_t)N_NODES * DIM;     // 50000*32
    float* tsum = tgt  + (size_t)N_NODES * DIM;     // 50000*32
    float* hb1  = tsum + (size_t)N_NODES * DIM;     // 64
    float* hb2  = hb1  + HID;                       // 32

    const int rowsBlocks = N_NODES / 8;             // 8 waves (rows) per 256-thread block
    const int gemmBlocks = (N_NODES / 16 + 7) / 8;  // 3125 waves, 8 waves/block

    hipMemsetAsync(tsum, 0, (size_t)N_NODES * DIM * sizeof(float), stream);
    k_expmap_in<<<rowsBlocks, 256, 0, stream>>>(x, xh, xn, N_NODES);

    for (int ib = 0; ib < NBRANCH; ib++) {
        const int*   er = k_rows + (size_t)ib * NEDGE;
        const int*   ec = k_cols + (size_t)ib * NEDGE;
        const float* ev = k_vals + (size_t)ib * NEDGE;

        // ---- layer 1 ----
        k_bias<HID><<<1, 32, 0, stream>>>(b1 + (size_t)ib * HID, hb1);
        k_gemm_wmma<FEAT, HID><<<gemmBlocks, 256, 0, stream>>>(
            xh, W1 + (size_t)ib * HID * FEAT, mx1, N_NODES);
        k_hyplin_fix<HID><<<rowsBlocks, 256, 0, stream>>>(mx1, xn, hb1, N_NODES);
        hipMemsetAsync(agg1, 0, (size_t)N_NODES * HID * sizeof(float), stream);
        {
            long th = (long)NEDGE * (HID / 4);
            k_spmm<HID><<<(int)((th + 255) / 256), 256, 0, stream>>>(er, ec, ev, mx1, agg1, NEDGE);
        }
        k_postagg1<<<rowsBlocks, 256, 0, stream>>>(agg1, h1, h1n, N_NODES);

        // ---- layer 2 ----
        k_bias<DIM><<<1, 32, 0, stream>>>(b2 + (size_t)ib * DIM, hb2);
        k_gemm_wmma<HID, DIM><<<gemmBlocks, 256, 0, stream>>>(
            h1, W2 + (size_t)ib * DIM * HID, mx2, N_NODES);
        k_hyplin_fix<DIM><<<rowsBlocks, 256, 0, stream>>>(mx2, h1n, hb2, N_NODES);
        hipMemsetAsync(agg2, 0, (size_t)N_NODES * DIM * sizeof(float), stream);
        {
            long th = (long)NEDGE * (DIM / 4);
            k_spmm<DIM><<<(int)((th + 255) / 256), 256, 0, stream>>>(er, ec, ev, mx2, agg2, NEDGE);
        }
        k_post2_combine<<<rowsBlocks, 256, 0, stream>>>(agg2, tgt, tsum, ib, N_NODES);
    }

    k_final<<<rowsBlocks, 256, 0, stream>>>(tsum, tgt, (float*)d_out, N_NODES);
}